// Graph_Encoder_Norm_32212254720636
// MI455X (gfx1250) — compile-verified
//
#include <hip/hip_runtime.h>
#include <math.h>

#define N_NODES 50000
#define N_EDGES 1600000
#define F_IN    256
#define DIM     64
#define L_LAYERS 4
#define GEPS    1e-5f

typedef __attribute__((ext_vector_type(2))) float v2f;
typedef __attribute__((ext_vector_type(8))) float v8f;

// ---------------- BatchNorm1d (eval) ----------------
__global__ void bn_kernel(const float* __restrict__ x,
                          const float* __restrict__ w, const float* __restrict__ b,
                          const float* __restrict__ mean, const float* __restrict__ var,
                          float* __restrict__ out) {
    int idx = blockIdx.x * blockDim.x + threadIdx.x;
    if (idx >= N_NODES * F_IN) return;
    int f = idx & (F_IN - 1);
    out[idx] = (x[idx] - mean[f]) * rsqrtf(var[f] + GEPS) * w[f] + b[f];
}

// ---------------- GEMM via V_WMMA_F32_16X16X4_F32 ----------------
// Block = 128 threads (4 waves). Block computes 16 rows x 64 cols of xw.
// Wave w computes the 16x16 tile at columns [16w, 16w+16).
__global__ __launch_bounds__(128)
void gemm_wmma_kernel(const float* __restrict__ A, const float* __restrict__ W,
                      float* __restrict__ out, int K) {
    int lane = threadIdx.x & 31;
    int wave = threadIdx.x >> 5;
    int half = lane >> 4;      // 0: lanes 0-15, 1: lanes 16-31
    int r    = lane & 15;
    int row0 = blockIdx.x * 16;
    int col0 = wave * 16;

    // A 16x4 f32 fragment: lane(half=0,r) holds (M=r, K=k..k+1); half=1 holds K=k+2..k+3
    const float* arow = A + (size_t)(row0 + r) * K + 2 * half;
    // B 4x16 f32 fragment: lane(half=0,r) holds (K=k..k+1, N=r); half=1 holds K=k+2..k+3
    const float* bcol = W + (size_t)(2 * half) * DIM + col0 + r;

    v8f c = {};
    for (int k = 0; k < K; k += 4) {
        v2f a;
        a.x = arow[k];
        a.y = arow[k + 1];
        v2f b;
        b.x = bcol[(size_t)k * DIM];
        b.y = bcol[(size_t)(k + 1) * DIM];
        c = __builtin_amdgcn_wmma_f32_16x16x4_f32(false, a, false, b,
                                                  (short)0, c, false, false);
    }
    // C/D layout: lane(half,r) VGPR j = (M = j + 8*half, N = r)
    float* orow = out + (size_t)(row0 + 8 * half) * DIM + col0 + r;
#pragma unroll
    for (int j = 0; j < 8; ++j) orow[(size_t)j * DIM] = c[j];
}

// ---------------- s = xw@a_src, t = xw@a_dst ----------------
__global__ void st_kernel(const float* __restrict__ xw, const float* __restrict__ asrc,
                          const float* __restrict__ adst, float* __restrict__ s,
                          float* __restrict__ t) {
    __shared__ float ls[256];
    __shared__ float lt[256];
    int tid  = threadIdx.x;
    int node = blockIdx.x * 4 + (tid >> 6);
    int f    = tid & 63;
    float v  = xw[(size_t)node * DIM + f];
    ls[tid] = v * asrc[f];
    lt[tid] = v * adst[f];
    __syncthreads();
    for (int off = 32; off > 0; off >>= 1) {
        if (f < off) { ls[tid] += ls[tid + off]; lt[tid] += lt[tid + off]; }
        __syncthreads();
    }
    if (f == 0) { s[node] = ls[tid]; t[node] = lt[tid]; }
}

// ---------------- per-layer scratch init ----------------
__global__ void init_kernel(float* __restrict__ acc, float* __restrict__ den,
                            float* __restrict__ m, float* __restrict__ fsum,
                            float* __restrict__ fvar) {
    int idx = blockIdx.x * blockDim.x + threadIdx.x;
    if (idx < N_NODES * DIM) acc[idx] = 0.f;
    if (idx < N_NODES) { den[idx] = 0.f; m[idx] = -INFINITY; }
    if (idx < DIM) { fsum[idx] = 0.f; fvar[idx] = 0.f; }
}

// float atomic max via signed/unsigned integer ordering trick
__device__ inline void atomicMaxFloat(float* addr, float val) {
    if (val >= 0.f) atomicMax((int*)addr, __float_as_int(val));
    else            atomicMin((unsigned int*)addr, __float_as_uint(val));
}

// ---------------- edge pass 1: logit + segment max ----------------
__global__ void edge_logit_kernel(const int* __restrict__ src, const int* __restrict__ dst,
                                  const float* __restrict__ s, const float* __restrict__ t,
                                  float* __restrict__ logit, float* __restrict__ m) {
    int e = blockIdx.x * blockDim.x + threadIdx.x;
    if (e >= N_EDGES) return;
    float v = s[src[e]] + t[dst[e]];
    v = v > 0.f ? v : 0.2f * v;          // leaky 0.2
    logit[e] = v;
    atomicMaxFloat(&m[dst[e]], v);
}

// ---------------- edge pass 2: e = exp(logit - m[dst]); den += e ----------------
__global__ void edge_exp_kernel(const int* __restrict__ dst, float* __restrict__ logit_eexp,
                                const float* __restrict__ m, float* __restrict__ den) {
    int e = blockIdx.x * blockDim.x + threadIdx.x;
    if (e >= N_EDGES) return;
    int d = dst[e];
    float ev = expf(logit_eexp[e] - m[d]);
    logit_eexp[e] = ev;                  // in-place reuse
    atomicAdd(&den[d], ev);
}

// ---------------- edge pass 3: acc[dst,:] += e * xw[src,:] ----------------
__global__ void edge_scatter_kernel(const int* __restrict__ src, const int* __restrict__ dst,
                                    const float* __restrict__ eexp,
                                    const float* __restrict__ xw,
                                    float* __restrict__ acc) {
    int idx = blockIdx.x * blockDim.x + threadIdx.x;   // < E*64 = 102.4M, fits int
    int e = idx >> 6;
    int f = idx & 63;
    float ev = eexp[e];
    atomicAdd(&acc[(size_t)dst[e] * DIM + f], ev * xw[(size_t)src[e] * DIM + f]);
}

// ---------------- normalize-by-den + bias, accumulate per-feature mean sums ----------------
__global__ void node_bias_sum_kernel(float* __restrict__ y /* acc in, y out (in place) */,
                                     const float* __restrict__ den,
                                     const float* __restrict__ bias,
                                     float* __restrict__ fsum) {
    __shared__ float lsum[DIM];
    int tid = threadIdx.x;
    if (tid < DIM) lsum[tid] = 0.f;
    __syncthreads();
    int idx = blockIdx.x * 256 + tid;
    int f   = tid & 63;
    int n   = idx >> 6;
    float d = den[n];
    float val = (d > 0.f ? y[idx] / d : 0.f) + bias[f];
    y[idx] = val;
    atomicAdd(&lsum[f], val);
    __syncthreads();
    if (tid < DIM) atomicAdd(&fsum[tid], lsum[tid]);
}

// ---------------- sh = y - ms*mean (in place), accumulate var sums ----------------
__global__ void node_var_kernel(float* __restrict__ y, const float* __restrict__ fsum,
                                const float* __restrict__ msc, float* __restrict__ fvar) {
    __shared__ float lvar[DIM];
    int tid = threadIdx.x;
    if (tid < DIM) lvar[tid] = 0.f;
    __syncthreads();
    int idx = blockIdx.x * 256 + tid;
    int f   = tid & 63;
    float mean = fsum[f] * (1.0f / (float)N_NODES);
    float sh = y[idx] - msc[f] * mean;
    y[idx] = sh;
    atomicAdd(&lvar[f], sh * sh);
    __syncthreads();
    if (tid < DIM) atomicAdd(&fvar[tid], lvar[tid]);
}

// ---------------- final: graphnorm scale + leaky(0.01), write xs/h/x ----------------
__global__ void node_final_kernel(const float* __restrict__ y, const float* __restrict__ fvar,
                                  const float* __restrict__ gw, const float* __restrict__ gb,
                                  float* __restrict__ xnext, float* __restrict__ h,
                                  float* __restrict__ out, int layer) {
    int idx = blockIdx.x * blockDim.x + threadIdx.x;
    if (idx >= N_NODES * DIM) return;
    int f = idx & 63;
    float var = fvar[f] * (1.0f / (float)N_NODES);
    float xn = gw[f] * y[idx] * rsqrtf(var + GEPS) + gb[f];
    xn = xn > 0.f ? xn : 0.01f * xn;     // leaky 0.01
    xnext[idx] = xn;
    // xs: shape [N*D, L] row-major -> element (idx, layer)
    out[(size_t)2 * N_NODES * DIM + (size_t)idx * L_LAYERS + layer] = xn;
    float hv = (layer == 0) ? 0.5f * xn : h[idx] + 0.5f * xn;
    h[idx] = hv;
    if (layer == L_LAYERS - 1) {
        out[idx] = xn;                          // x
        out[(size_t)N_NODES * DIM + idx] = hv;  // h
    }
}

extern "C" void kernel_launch(void* const* d_in, const int* in_sizes, int n_in,
                              void* d_out, int out_size, void* d_ws, size_t ws_size,
                              hipStream_t stream) {
    const float* x       = (const float*)d_in[0];
    const int*   ei      = (const int*)  d_in[1];
    const float* bnw     = (const float*)d_in[2];
    const float* bnb     = (const float*)d_in[3];
    const float* bnm     = (const float*)d_in[4];
    const float* bnv     = (const float*)d_in[5];
    const float* W0      = (const float*)d_in[6];
    const float* Ws      = (const float*)d_in[7];
    const float* att_src = (const float*)d_in[8];
    const float* att_dst = (const float*)d_in[9];
    const float* bias    = (const float*)d_in[10];
    const float* gnw     = (const float*)d_in[11];
    const float* gnb     = (const float*)d_in[12];
    const float* gns     = (const float*)d_in[13];
    float* out = (float*)d_out;

    const int* src = ei;              // edge_index[0]
    const int* dst = ei + N_EDGES;    // edge_index[1]

    // workspace layout (floats)
    float* ws  = (float*)d_ws;
    float* XBN = ws;                                   // N*F
    float* XW  = XBN + (size_t)N_NODES * F_IN;         // N*D
    float* Y   = XW  + (size_t)N_NODES * DIM;          // N*D  (acc -> y -> sh, in place)
    float* XN  = Y   + (size_t)N_NODES * DIM;          // N*D  next-layer input
    float* H   = XN  + (size_t)N_NODES * DIM;          // N*D  running h
    float* Sv  = H   + (size_t)N_NODES * DIM;          // N
    float* Tv  = Sv  + N_NODES;                        // N
    float* Mv  = Tv  + N_NODES;                        // N
    float* DEN = Mv  + N_NODES;                        // N
    float* LG  = DEN + N_NODES;                        // E (logit, then e)
    float* FS  = LG  + N_EDGES;                        // 64
    float* FV  = FS  + DIM;                            // 64

    bn_kernel<<<(N_NODES * F_IN) / 256, 256, 0, stream>>>(x, bnw, bnb, bnm, bnv, XBN);

    for (int l = 0; l < L_LAYERS; ++l) {
        const float* A = (l == 0) ? XBN : XN;
        const float* W = (l == 0) ? W0 : (Ws + (size_t)(l - 1) * DIM * DIM);
        int K = (l == 0) ? F_IN : DIM;

        gemm_wmma_kernel<<<N_NODES / 16, 128, 0, stream>>>(A, W, XW, K);
        st_kernel<<<N_NODES / 4, 256, 0, stream>>>(XW, att_src + l * DIM, att_dst + l * DIM, Sv, Tv);
        init_kernel<<<(N_NODES * DIM) / 256, 256, 0, stream>>>(Y, DEN, Mv, FS, FV);
        edge_logit_kernel<<<N_EDGES / 256, 256, 0, stream>>>(src, dst, Sv, Tv, LG, Mv);
        edge_exp_kernel<<<N_EDGES / 256, 256, 0, stream>>>(dst, LG, Mv, DEN);
        edge_scatter_kernel<<<(N_EDGES / 256) * DIM, 256, 0, stream>>>(src, dst, LG, XW, Y);
        node_bias_sum_kernel<<<(N_NODES * DIM) / 256, 256, 0, stream>>>(Y, DEN, bias + l * DIM, FS);
        node_var_kernel<<<(N_NODES * DIM) / 256, 256, 0, stream>>>(Y, FS, gns + l * DIM, FV);
        node_final_kernel<<<(N_NODES * DIM) / 256, 256, 0, stream>>>(
            Y, FV, gnw + l * DIM, gnb + l * DIM, XN, H, out, l);
    }
}